// SPDNet_43319040148099
// MI455X (gfx1250) — compile-verified
//
#include <hip/hip_runtime.h>
#include <hip/hip_bf16.h>
#include <math.h>

// ---------------- WMMA f32 16x16x4 helpers (CDNA5 / gfx1250, wave32) --------
typedef __attribute__((ext_vector_type(2))) float v2f;
typedef __attribute__((ext_vector_type(8))) float v8f;
typedef __attribute__((__vector_size__(4 * sizeof(int)))) int v4i;

#define AS1 __attribute__((address_space(1)))
#define AS3 __attribute__((address_space(3)))

#if defined(__gfx1250__) &&                                            \
    __has_builtin(__builtin_amdgcn_global_load_async_to_lds_b32) &&    \
    __has_builtin(__builtin_amdgcn_global_load_async_to_lds_b128)
#define HAS_ASYNC 1
#else
#define HAS_ASYNC 0
#endif

// Async global->LDS copy of 4 consecutive floats (16B aligned both sides).
__device__ __forceinline__ void a_copy4(const float* g, float* l) {
#if HAS_ASYNC
    __builtin_amdgcn_global_load_async_to_lds_b128(
        (AS1 v4i*)(void*)g, (AS3 v4i*)(void*)l, 0, 0);
#else
    l[0] = g[0]; l[1] = g[1]; l[2] = g[2]; l[3] = g[3];
#endif
}
// Async global->LDS copy of 1 float.
__device__ __forceinline__ void a_copy1(const float* g, float* l) {
#if HAS_ASYNC
    __builtin_amdgcn_global_load_async_to_lds_b32(
        (AS1 int*)(void*)g, (AS3 int*)(void*)l, 0, 0);
#else
    l[0] = g[0];
#endif
}
__device__ __forceinline__ void wait_async() {
#if HAS_ASYNC
#if __has_builtin(__builtin_amdgcn_s_wait_asynccnt)
    __builtin_amdgcn_s_wait_asynccnt(0);
#else
    asm volatile("s_wait_asynccnt 0x0" ::: "memory");
#endif
#endif
}

__device__ __forceinline__ v8f fzero() {
    v8f z;
#pragma unroll
    for (int e = 0; e < 8; ++e) z[e] = 0.0f;
    return z;
}

__device__ __forceinline__ v8f wmma4(v2f a, v2f b, v8f c) {
    // D = A(16x4,f32) * B(4x16,f32) + C(16x16,f32)
    return __builtin_amdgcn_wmma_f32_16x16x4_f32(
        /*neg_a=*/false, a, /*neg_b=*/false, b,
        /*c_mod=*/(short)0, c, /*reuse_a=*/false, /*reuse_b=*/false);
}

// A-matrix 16x4 fragment (ISA 7.12.2): lanes 0-15 hold K=0,1; lanes 16-31 K=2,3
__device__ __forceinline__ v2f load_a_frag(const float* A, int lda, int row0, int k0) {
    int l = threadIdx.x & 31;
    int m = l & 15, h = l >> 4;
    const float* p = A + (size_t)(row0 + m) * lda + k0 + 2 * h;
    v2f r; r.x = p[0]; r.y = p[1];
    return r;
}

// A = W^T fragment: A[m][k] = W[k][m]
__device__ __forceinline__ v2f load_at_frag(const float* W, int ldw, int row0, int k0) {
    int l = threadIdx.x & 31;
    int m = l & 15, h = l >> 4;
    const float* p = W + (size_t)(k0 + 2 * h) * ldw + row0 + m;
    v2f r; r.x = p[0]; r.y = p[ldw];
    return r;
}

// B-matrix 4x16 fragment: lanes 0-15 hold K=0,1 (as .x,.y); lanes 16-31 K=2,3
__device__ __forceinline__ v2f load_b_frag(const float* B, int ldb, int k0, int col0) {
    int l = threadIdx.x & 31;
    int n = l & 15, h = l >> 4;
    const float* p = B + (size_t)(k0 + 2 * h) * ldb + col0 + n;
    v2f r; r.x = p[0]; r.y = p[ldb];
    return r;
}

// C/D 16x16: VGPR r -> row r (lanes 0-15) / row r+8 (lanes 16-31)
__device__ __forceinline__ void store_d(float* C, int ldc, int row0, int col0, v8f d) {
    int l = threadIdx.x & 31;
    int n = l & 15, h = l >> 4;
#pragma unroll
    for (int r = 0; r < 8; ++r)
        C[(size_t)(row0 + r + (h ? 8 : 0)) * ldc + col0 + n] = d[r];
}

// ---------------- Kernel 1: column-normalize W1 (128x64) and W2 (64x64) -----
__global__ void normalize_kernel(const float* __restrict__ W1,
                                 const float* __restrict__ W2,
                                 float* __restrict__ W1n,
                                 float* __restrict__ W2n) {
    int j = threadIdx.x;
    if (j >= 64) return;
    if (blockIdx.x == 0) {
        float s = 0.f;
        for (int i = 0; i < 128; ++i) { float w = W1[i * 64 + j]; s += w * w; }
        float inv = 1.f / (sqrtf(s) + 1e-8f);
        for (int i = 0; i < 128; ++i) W1n[i * 64 + j] = W1[i * 64 + j] * inv;
    } else {
        float s = 0.f;
        for (int i = 0; i < 64; ++i) { float w = W2[i * 64 + j]; s += w * w; }
        float inv = 1.f / (sqrtf(s) + 1e-8f);
        for (int i = 0; i < 64; ++i) W2n[i * 64 + j] = W2[i * 64 + j] * inv;
    }
}

// ---------------- Kernel 2: Y1 = W1n^T X W1n  (per batch, WMMA) -------------
// 256 threads = 8 waves per workgroup; one workgroup per batch matrix.
// Strides chosen so async-B128 LDS targets are 16B aligned AND fragment reads
// are bank-conflict-free (20, 36, 68 are 4-float multiples with odd /4).
__global__ __launch_bounds__(256) void bimap1_kernel(const float* __restrict__ X,
                                                     const float* __restrict__ W1n,
                                                     float* __restrict__ Y1) {
    const int b = blockIdx.x;
    const float* Xb = X + (size_t)b * 128 * 128;
    __shared__ float Xp[128][20];  // K-panel of X (128 rows x 16 k), async dest
    __shared__ float Wp[16][68];   // K-panel of W1n (16 k x 64 cols), async dest
    __shared__ float Ps[128][65];  // P = X*W1n (128x64), written by store_d
    const int t = threadIdx.x;
    const int wave = t >> 5;

    // ---- Step 1: P = X * W1n, accumulate over 8 K-panels of 16 ----
    v8f acc[4];
#pragma unroll
    for (int i = 0; i < 4; ++i) acc[i] = fzero();

    const int mrow = wave * 16;  // each wave owns one 16-row stripe of P
    for (int kp = 0; kp < 8; ++kp) {
        const int k0 = kp * 16;
        // async stage X[:, k0:k0+16] : 512 float4 chunks
#pragma unroll
        for (int c = t; c < 512; c += 256) {
            int i = c >> 2, j4 = (c & 3) * 4;
            a_copy4(Xb + (size_t)i * 128 + k0 + j4, &Xp[i][j4]);
        }
        {   // async stage W1n[k0:k0+16, :] : 256 float4 chunks (1/thread)
            int row = t >> 4, j4 = (t & 15) * 4;
            a_copy4(W1n + (size_t)(k0 + row) * 64 + j4, &Wp[row][j4]);
        }
        wait_async();
        __syncthreads();

#pragma unroll
        for (int nt = 0; nt < 4; ++nt) {
#pragma unroll
            for (int kk = 0; kk < 16; kk += 4) {
                v2f a = load_a_frag(&Xp[0][0], 20, mrow, kk);
                v2f bf = load_b_frag(&Wp[0][0], 68, kk, nt * 16);
                acc[nt] = wmma4(a, bf, acc[nt]);
            }
        }
        __syncthreads();
    }
#pragma unroll
    for (int nt = 0; nt < 4; ++nt) store_d(&Ps[0][0], 65, mrow, nt * 16, acc[nt]);
    __syncthreads();

    // ---- Step 2: Y1 = W1n^T * P  (64x64), 16 tiles over 8 waves ----
    float* Y1b = Y1 + (size_t)b * 64 * 64;
#pragma unroll
    for (int i = 0; i < 2; ++i) {
        int tile = wave * 2 + i;
        int mt = tile >> 2, nt = tile & 3;
        v8f c = fzero();
        for (int kk = 0; kk < 128; kk += 4) {
            v2f a = load_at_frag(W1n, 64, mt * 16, kk);     // A = W1n^T
            v2f bf = load_b_frag(&Ps[0][0], 65, kk, nt * 16);
            c = wmma4(a, bf, c);
        }
        store_d(Y1b, 64, mt * 16, nt * 16, c);
    }
}

// ---------------- Kernel 3: in-place eigh (cyclic Jacobi) + log reconstruct -
// Fuses log_map(re_eig(M)) = V diag(log(max(w,1e-4))) V^T.
// One 64-thread workgroup (2 waves) per 64x64 symmetric matrix.
__global__ __launch_bounds__(64) void jacobi_log_kernel(float* __restrict__ M) {
    const int b = blockIdx.x;
    float* G = M + (size_t)b * 64 * 64;
    __shared__ float A[64][65];
    __shared__ float V[64][65];
    __shared__ float cs[32], sn[32], fv[64];
    __shared__ int pp[32], qq[32];
    const int t = threadIdx.x;  // 0..63 : owns column t (row phase) / row t (col phase)

    for (int i = 0; i < 64; ++i) {
        a_copy1(G + (size_t)i * 64 + t, &A[i][t]);  // async b32, coalesced per row
        V[i][t] = (i == t) ? 1.f : 0.f;
    }
    wait_async();
    __syncthreads();

    const int NSWEEP = 9;
    for (int sw = 0; sw < NSWEEP; ++sw) {
        for (int r = 0; r < 63; ++r) {
            // 32 disjoint pairs per round (round-robin tournament, 63 fixed)
            if (t < 32) {
                int p, q;
                if (t == 0) { p = 63; q = r % 63; }
                else { p = (r + t) % 63; q = (r + 63 - t) % 63; }
                if (p > q) { int x = p; p = q; q = x; }
                float app = A[p][p], aqq = A[q][q], apq = A[p][q];
                float c = 1.f, s = 0.f;
                if (fabsf(apq) > 1e-12f) {
                    float tau = (aqq - app) / (2.f * apq);
                    float tt = 1.f / (fabsf(tau) + sqrtf(1.f + tau * tau));
                    tt = (tau >= 0.f) ? tt : -tt;
                    c = rsqrtf(1.f + tt * tt);
                    s = tt * c;
                }
                cs[t] = c; sn[t] = s; pp[t] = p; qq[t] = q;
            }
            __syncthreads();
            // Row phase: A <- G^T A   (thread t updates column t of rows p,q)
#pragma unroll 4
            for (int k = 0; k < 32; ++k) {
                int p = pp[k], q = qq[k];
                float c = cs[k], s = sn[k];
                float ap = A[p][t], aq = A[q][t];
                A[p][t] = c * ap - s * aq;
                A[q][t] = s * ap + c * aq;
            }
            __syncthreads();
            // Col phase: A <- A G ; V <- V G  (thread t updates row t)
#pragma unroll 4
            for (int k = 0; k < 32; ++k) {
                int p = pp[k], q = qq[k];
                float c = cs[k], s = sn[k];
                float ap = A[t][p], aq = A[t][q];
                A[t][p] = c * ap - s * aq;
                A[t][q] = s * ap + c * aq;
                float vp = V[t][p], vq = V[t][q];
                V[t][p] = c * vp - s * vq;
                V[t][q] = s * vp + c * vq;
            }
            __syncthreads();
        }
    }

    fv[t] = logf(fmaxf(A[t][t], 1e-4f));
    __syncthreads();

    // Z = V diag(fv) V^T  via WMMA f32; 16 tiles over 2 waves, write back to G
    const int wave = t >> 5;
    const int l = t & 31, n = l & 15, h = l >> 4;
    for (int i = 0; i < 8; ++i) {
        int tile = wave * 8 + i;
        int mt = tile >> 2, nt = tile & 3;
        v8f c = fzero();
        for (int kk = 0; kk < 64; kk += 4) {
            v2f a = load_a_frag(&V[0][0], 65, mt * 16, kk);
            int k = kk + 2 * h;
            v2f bf;  // B[k][n] = fv[k] * V^T[k][n] = fv[k] * V[nt*16+n][k]
            bf.x = fv[k] * V[nt * 16 + n][k];
            bf.y = fv[k + 1] * V[nt * 16 + n][k + 1];
            c = wmma4(a, bf, c);
        }
        store_d(G, 64, mt * 16, nt * 16, c);
    }
}

// ---------------- Kernel 4: Y2 = W2n^T Z W2n (per batch, WMMA) --------------
__global__ __launch_bounds__(128) void bimap2_kernel(const float* __restrict__ Z,
                                                     const float* __restrict__ W2n,
                                                     float* __restrict__ Y2) {
    const int b = blockIdx.x;
    const float* Zb = Z + (size_t)b * 4096;
    __shared__ float Zs[64][68];   // async dest (16B-aligned rows)
    __shared__ float Ws[64][68];   // async dest
    __shared__ float Ts[64][65];   // written by store_d
    const int t = threadIdx.x;  // 0..127 (4 waves)
#pragma unroll
    for (int c = t; c < 1024; c += 128) {   // 64x64 = 1024 float4 chunks each
        int row = c >> 4, j4 = (c & 15) * 4;
        a_copy4(Zb + (size_t)row * 64 + j4, &Zs[row][j4]);
        a_copy4(W2n + (size_t)row * 64 + j4, &Ws[row][j4]);
    }
    wait_async();
    __syncthreads();
    const int wave = t >> 5;
    // T = W2n^T * Z
#pragma unroll
    for (int i = 0; i < 4; ++i) {
        int tile = wave * 4 + i, mt = tile >> 2, nt = tile & 3;
        v8f c = fzero();
        for (int kk = 0; kk < 64; kk += 4) {
            v2f a = load_at_frag(&Ws[0][0], 68, mt * 16, kk);
            v2f bf = load_b_frag(&Zs[0][0], 68, kk, nt * 16);
            c = wmma4(a, bf, c);
        }
        store_d(&Ts[0][0], 65, mt * 16, nt * 16, c);
    }
    __syncthreads();
    // Y2 = T * W2n
#pragma unroll
    for (int i = 0; i < 4; ++i) {
        int tile = wave * 4 + i, mt = tile >> 2, nt = tile & 3;
        v8f c = fzero();
        for (int kk = 0; kk < 64; kk += 4) {
            v2f a = load_a_frag(&Ts[0][0], 65, mt * 16, kk);
            v2f bf = load_b_frag(&Ws[0][0], 68, kk, nt * 16);
            c = wmma4(a, bf, c);
        }
        store_d(Y2 + (size_t)b * 4096, 64, mt * 16, nt * 16, c);
    }
}

// ---------------- Kernel 5: FC out = Z2_flat @ fc_w^T + fc_b ----------------
// M=4096 (batch), N=130 (pad to 144), K=4096. grid=(32, 9), 256 thr / block.
__global__ __launch_bounds__(256) void fc_kernel(const float* __restrict__ Z2,
                                                 const float* __restrict__ fcw,
                                                 const float* __restrict__ fcb,
                                                 float* __restrict__ out) {
    const int row0 = blockIdx.x * 128;
    const int col0 = blockIdx.y * 16;
    __shared__ float Zp[128][36];  // async dest (stride 36 floats: 16B-aligned, conflict-free)
    __shared__ float Wp[32][17];   // transposed panel, async b32
    const int t = threadIdx.x;
    const int wave = t >> 5;
    v8f acc = fzero();
    const int mrow = wave * 16;
    const bool full_tile = (col0 + 16 <= 130);   // uniform per block

    for (int kp = 0; kp < 128; ++kp) {
        const int k0 = kp * 32;
        // Zp[128][32] : 1024 float4 chunks
#pragma unroll
        for (int c = t; c < 1024; c += 256) {
            int row = c >> 3, j4 = (c & 7) * 4;
            a_copy4(Z2 + (size_t)(row0 + row) * 4096 + k0 + j4, &Zp[row][j4]);
        }
        // Wp[kk][n] = fc_w[col0+n][k0+kk]  (transposed; element-wise)
        if (full_tile) {
#pragma unroll
            for (int e = t; e < 512; e += 256) {
                int kk = e >> 4, n = e & 15;
                a_copy1(fcw + (size_t)(col0 + n) * 4096 + k0 + kk, &Wp[kk][n]);
            }
        } else {
#pragma unroll
            for (int e = t; e < 512; e += 256) {
                int kk = e >> 4, n = e & 15;
                int cc = col0 + n;
                Wp[kk][n] = (cc < 130) ? fcw[(size_t)cc * 4096 + k0 + kk] : 0.f;
            }
        }
        wait_async();
        __syncthreads();
#pragma unroll
        for (int kk = 0; kk < 32; kk += 4) {
            v2f a = load_a_frag(&Zp[0][0], 36, mrow, kk);
            v2f bf = load_b_frag(&Wp[0][0], 17, kk, 0);
            acc = wmma4(a, bf, acc);
        }
        __syncthreads();
    }
    {   // store with bias; guard N to 130
        int l = t & 31, n = l & 15, h = l >> 4;
        int col = col0 + n;
        if (col < 130) {
            float bias = fcb[col];
            int r0 = row0 + mrow + (h ? 8 : 0);
#pragma unroll
            for (int r = 0; r < 8; ++r)
                out[(size_t)(r0 + r) * 130 + col] = acc[r] + bias;
        }
    }
}

// ---------------- Host-side launch ------------------------------------------
extern "C" void kernel_launch(void* const* d_in, const int* in_sizes, int n_in,
                              void* d_out, int out_size, void* d_ws, size_t ws_size,
                              hipStream_t stream) {
    const float* X    = (const float*)d_in[0];  // 4096 x 128 x 128
    const float* W1   = (const float*)d_in[1];  // 128 x 64
    const float* W2   = (const float*)d_in[2];  // 64 x 64
    const float* fc_w = (const float*)d_in[3];  // 130 x 4096
    const float* fc_b = (const float*)d_in[4];  // 130
    float* out = (float*)d_out;                 // 4096 x 130
    (void)in_sizes; (void)n_in; (void)out_size; (void)ws_size;

    const int B = 4096;
    char* ws = (char*)d_ws;
    size_t off = 0;
    auto carve = [&](size_t bytes) {
        void* p = ws + off;
        off = (off + bytes + 255) & ~(size_t)255;
        return p;
    };
    float* W1n  = (float*)carve(128 * 64 * sizeof(float));
    float* W2n  = (float*)carve(64 * 64 * sizeof(float));
    float* bufA = (float*)carve((size_t)B * 64 * 64 * sizeof(float));  // Y1 -> Z1
    float* bufB = (float*)carve((size_t)B * 64 * 64 * sizeof(float));  // Y2 -> Z2

    normalize_kernel<<<2, 64, 0, stream>>>(W1, W2, W1n, W2n);
    bimap1_kernel<<<B, 256, 0, stream>>>(X, W1n, bufA);
    jacobi_log_kernel<<<B, 64, 0, stream>>>(bufA);        // Z1 in place
    bimap2_kernel<<<B, 128, 0, stream>>>(bufA, W2n, bufB);
    jacobi_log_kernel<<<B, 64, 0, stream>>>(bufB);        // Z2 in place
    fc_kernel<<<dim3(32, 9), 256, 0, stream>>>(bufB, fc_w, fc_b, out);
}